// heterogeneousGNN_72516227826099
// MI455X (gfx1250) — compile-verified
//
#include <hip/hip_runtime.h>
#include <hip/hip_bf16.h>
#include <math.h>
#include <stdint.h>

#define N_SLOT   30000
#define N_INT    20000
#define N_NODES  50000
#define DIM      128
#define N_REL    16
#define EPSF     1e-15f
#define CLAMPF   1e-7f

typedef int v4i __attribute__((vector_size(16)));

// CDNA5 async global->LDS copy of one 16B chunk per lane (global_load_async_to_lds_b128)
__device__ __forceinline__ void async_copy_b128(const float* gsrc, float* lds_dst) {
#if __has_builtin(__builtin_amdgcn_global_load_async_to_lds_b128)
    __attribute__((address_space(1))) v4i* gp =
        (__attribute__((address_space(1))) v4i*)(uintptr_t)gsrc;
    __attribute__((address_space(3))) float* lp0 =
        (__attribute__((address_space(3))) float*)lds_dst;
    __attribute__((address_space(3))) v4i* lp =
        (__attribute__((address_space(3))) v4i*)lp0;
    __builtin_amdgcn_global_load_async_to_lds_b128(gp, lp, 0, 0);
#else
    const float4 v = *(const float4*)gsrc;
    *(float4*)lds_dst = v;
#endif
}

__device__ __forceinline__ void wait_async() {
#if __has_builtin(__builtin_amdgcn_s_wait_asynccnt)
    __builtin_amdgcn_s_wait_asynccnt(0);
#else
    asm volatile("s_wait_asynccnt 0x0" ::: "memory");
#endif
}

__device__ __forceinline__ float dot4(float ax, float ay, float az, float aw,
                                      float bx, float by, float bz, float bw) {
    return ax * bx + ay * by + az * bz + aw * bw;
}

// wave32 butterfly reduction: every lane ends with the full 32-lane sum
__device__ __forceinline__ float wave_sum(float v) {
    v += __shfl_xor(v, 16, 32);
    v += __shfl_xor(v, 8, 32);
    v += __shfl_xor(v, 4, 32);
    v += __shfl_xor(v, 2, 32);
    v += __shfl_xor(v, 1, 32);
    return v;
}

__global__ void zero_kernel(float* __restrict__ out, float* __restrict__ cnt,
                            int n_out, int n_cnt) {
    const int tid = blockIdx.x * blockDim.x + threadIdx.x;
    const int stride = gridDim.x * blockDim.x;
    for (int i = tid; i < n_out; i += stride) out[i] = 0.0f;
    for (int i = tid; i < n_cnt; i += stride) cnt[i] = 0.0f;
}

__global__ __launch_bounds__(256) void edge_kernel(
    const float* __restrict__ slot_embed,   // [N_SLOT, 128]
    const float* __restrict__ int_embed,    // [N_INT, 128]
    const float* __restrict__ rel_embed,    // [16, 128]
    const int*   __restrict__ src,
    const int*   __restrict__ dst,
    const int*   __restrict__ etype,
    float* __restrict__ sums,               // [N_NODES, 128] accumulator (d_out)
    float* __restrict__ cnt,                // [N_NODES]
    int n_edges)
{
    __shared__ float s_rel[N_REL * DIM];    // 8 KB relation table

    // ---- stage relation table into LDS via CDNA5 async global->LDS path ----
    {
        const int tid = threadIdx.x;
        // 512 x b128 transfers; consecutive lanes copy consecutive 16B chunks.
        for (int i = tid; i < (N_REL * DIM) / 4; i += 256) {
            async_copy_b128(rel_embed + i * 4, &s_rel[i * 4]);
        }
        wait_async();
    }
    __syncthreads();

    const int lane = threadIdx.x & 31;
    const int wid  = threadIdx.x >> 5;
    const int edge = blockIdx.x * 8 + wid;
    if (edge >= n_edges) return;

    const int s = src[edge];
    const int d = dst[edge];
    const int e = etype[edge];

    const float* srow = (s < N_SLOT) ? (slot_embed + (size_t)s * DIM)
                                     : (int_embed + (size_t)(s - N_SLOT) * DIM);
    const float* drow = (d < N_SLOT) ? (slot_embed + (size_t)d * DIM)
                                     : (int_embed + (size_t)(d - N_SLOT) * DIM);

    // 1 wave = 1 edge; 4 contiguous floats/lane -> global_load_b128, 512B/row
    const float4 es = *(const float4*)(srow + lane * 4);
    const float4 ed = *(const float4*)(drow + lane * 4);
    const float4 rl = *(const float4*)(&s_rel[e * DIM + lane * 4]);

    // ---- node1[src] = expmap0(embed[src]) ----
    const float n2 = wave_sum(dot4(es.x, es.y, es.z, es.w, es.x, es.y, es.z, es.w));
    const float n  = sqrtf(fmaxf(n2, EPSF));
    const float th = tanhf(n);
    const float sc = th / n;
    const float xx = es.x * sc, xy_ = es.y * sc, xz = es.z * sc, xw = es.w * sc;
    const float x2 = th * th;                   // sqnorm(node1[src])

    // ---- tangent = embed[dst] + rel[etype]; u = ptransp0(x, tangent) ----
    const float tx = ed.x + rl.x, ty = ed.y + rl.y, tz = ed.z + rl.z, tw = ed.w + rl.w;
    const float one_m = 1.0f - x2;
    const float ux = tx * one_m, uy = ty * one_m, uz = tz * one_m, uw = tw * one_m;

    // ---- expmap(u, x): second = tanh(0.5*lam*|u|) * u / |u| ----
    const float un2 = wave_sum(dot4(ux, uy, uz, uw, ux, uy, uz, uw));
    const float un  = sqrtf(fmaxf(un2, EPSF));
    const float lam = 2.0f / fmaxf(one_m, EPSF);
    const float ys  = tanhf(0.5f * lam * un) / un;
    const float yx = ux * ys, yy = uy * ys, yz = uz * ys, yw = uw * ys;

    // ---- mobius_add(x, second) ----
    const float y2 = wave_sum(dot4(yx, yy, yz, yw, yx, yy, yz, yw));
    const float xy = wave_sum(dot4(xx, xy_, xz, xw, yx, yy, yz, yw));
    const float ca = 1.0f + 2.0f * xy + y2;     // coeff on x
    const float cb = one_m;                     // coeff on second
    const float den = 1.0f + 2.0f * xy + x2 * y2;
    const float rden = 1.0f / fmaxf(den, EPSF);
    const float qx = (ca * xx + cb * yx) * rden;
    const float qy = (ca * xy_ + cb * yy) * rden;
    const float qz = (ca * xz + cb * yz) * rden;
    const float qw = (ca * xw + cb * yw) * rden;

    // ---- logmap0(q) = artanh(|q|) * q / |q| ----
    const float q2 = wave_sum(dot4(qx, qy, qz, qw, qx, qy, qz, qw));
    const float qn = sqrtf(fmaxf(q2, EPSF));
    const float a  = fminf(qn, 1.0f - CLAMPF);          // qn >= 0
    const float at = 0.5f * (log1pf(a) - log1pf(-a));   // artanh
    const float ts = at / qn;

    // ---- segment-sum scatter onto src node ----
    float* o = sums + (size_t)s * DIM + lane * 4;
    atomicAdd(o + 0, qx * ts);
    atomicAdd(o + 1, qy * ts);
    atomicAdd(o + 2, qz * ts);
    atomicAdd(o + 3, qw * ts);
    if (lane == 0) atomicAdd(cnt + s, 1.0f);
}

__global__ void finalize_kernel(float* __restrict__ out,
                                const float* __restrict__ cnt, int n_out) {
    const int i = blockIdx.x * blockDim.x + threadIdx.x;
    if (i < n_out) {
        const float c = fmaxf(cnt[i >> 7], 1.0f);   // 128 elems per node
        out[i] = out[i] / c;
    }
}

extern "C" void kernel_launch(void* const* d_in, const int* in_sizes, int n_in,
                              void* d_out, int out_size, void* d_ws, size_t ws_size,
                              hipStream_t stream) {
    (void)n_in; (void)ws_size;
    const float* slot_embed = (const float*)d_in[0];
    const float* int_embed  = (const float*)d_in[1];
    const float* rel_embed  = (const float*)d_in[2];
    const int*   src        = (const int*)d_in[3];
    const int*   dst        = (const int*)d_in[4];
    const int*   etype      = (const int*)d_in[5];

    float* out = (float*)d_out;           // [50000 * 128] sums -> means
    float* cnt = (float*)d_ws;            // [50000] edge counts

    const int n_edges = in_sizes[3];
    const int n_out   = out_size;

    zero_kernel<<<1024, 256, 0, stream>>>(out, cnt, n_out, N_NODES);

    const int blocks = (n_edges + 7) / 8; // 8 waves/block, 1 edge/wave
    edge_kernel<<<blocks, 256, 0, stream>>>(slot_embed, int_embed, rel_embed,
                                            src, dst, etype, out, cnt, n_edges);

    finalize_kernel<<<(n_out + 255) / 256, 256, 0, stream>>>(out, cnt, n_out);
}